// DIETModel_66769561584247
// MI455X (gfx1250) — compile-verified
//
#include <hip/hip_runtime.h>
#include <math.h>

typedef __attribute__((ext_vector_type(2))) float v2f;
typedef __attribute__((ext_vector_type(8))) float v8f;

#define WORDN  30000
#define FDIM   80000
#define Bsz    32
#define Lsz    64
#define Kgr    6
#define Dm     256
#define Hh     8
#define FFDN   512
#define NLAYERS 2
#define NTOK   (Bsz*Lsz)   /* 2048 */
#define EPSV   1e-5f

#define LPAD   36          /* LDS row pitch in floats: 144B, 16B aligned */

__device__ __forceinline__ v8f wmma4(v2f a, v2f b, v8f c) {
  // V_WMMA_F32_16X16X4_F32 : D = A(16x4) * B(4x16) + C(16x16)
  return __builtin_amdgcn_wmma_f32_16x16x4_f32(false, a, false, b, (short)0, c,
                                               false, false);
}

// Async global -> LDS copy of 16B per lane (gfx1250, tracked by ASYNCcnt).
__device__ __forceinline__ void async_copy_b128(unsigned lds_off,
                                                unsigned g_off,
                                                unsigned long long g_base) {
  asm volatile("global_load_async_to_lds_b128 %0, %1, %2"
               :
               : "v"(lds_off), "v"(g_off), "s"(g_base)
               : "memory");
}

__device__ __forceinline__ void wait_async0() {
  asm volatile("s_wait_asynccnt 0x0" ::: "memory");
}

// ---------------- sparse embedding: h0 = relu(multi_hot @ fc1_w^T + b) -------
// Fixed 7-way gather with 0/1 dedupe weights so all 7 loads issue in parallel.
__global__ __launch_bounds__(256)
void gather_fc1_kernel(const int* __restrict__ word_idx,
                       const int* __restrict__ ngram_idx,
                       const float* __restrict__ w,
                       const float* __restrict__ bias,
                       float* __restrict__ h0) {
  int tok = blockIdx.x;
  int   f[7];
  float wt[7];
  f[0]  = word_idx[tok];
  wt[0] = 1.f;
  for (int i = 0; i < Kgr; ++i) {
    int fi = WORDN + ngram_idx[tok * Kgr + i];
    bool dup = false;
    for (int j = 1; j <= i; ++j) dup |= (f[j] == fi);   // multi-hot: dedupe
    f[i + 1]  = fi;
    wt[i + 1] = dup ? 0.f : 1.f;
  }
  for (int j = threadIdx.x; j < FFDN; j += blockDim.x) {
    const float* row = w + (size_t)j * FDIM;
    float s = bias[j];
#pragma unroll
    for (int i = 0; i < 7; ++i) s += wt[i] * row[f[i]];
    h0[(size_t)tok * FFDN + j] = fmaxf(s, 0.f);
  }
}

// ---------------- WMMA GEMM: C[M,N] = A[M,K] * W[N,K]^T + bias ---------------
// Block = 256 threads = 8 waves; block tile 64(M) x 128(N); K-blocks of 32
// double-buffered in LDS via async global->LDS copies.
// grid = (M/64, N/128); K must be a multiple of 32.
__global__ __launch_bounds__(256)
void gemm_wmma_kernel(const float* __restrict__ A, const float* __restrict__ W,
                      const float* __restrict__ bias, float* __restrict__ C,
                      int M, int N, int K, int relu) {
  (void)M;
  __shared__ __align__(16) float sA[2][64][LPAD];
  __shared__ __align__(16) float sW[2][128][LPAD];

  int tid  = threadIdx.x;
  int lane = tid & 31;
  int wave = tid >> 5;
  int lm   = lane & 15;
  int half = lane >> 4;
  int rowBase = blockIdx.x * 64;
  int colBase = blockIdx.y * 128;
  int wm = (wave & 3) * 16;    // wave's row tile inside block
  int wn = (wave >> 2) * 64;   // wave's 64-col strip inside block

  unsigned ldsA[2], ldsW[2];
  ldsA[0] = (unsigned)(uintptr_t)&sA[0][0][0];
  ldsA[1] = (unsigned)(uintptr_t)&sA[1][0][0];
  ldsW[0] = (unsigned)(uintptr_t)&sW[0][0][0];
  ldsW[1] = (unsigned)(uintptr_t)&sW[1][0][0];
  unsigned long long aBase = (unsigned long long)(uintptr_t)A;
  unsigned long long wBase = (unsigned long long)(uintptr_t)W;

  v8f acc[4];
  acc[0] = {}; acc[1] = {}; acc[2] = {}; acc[3] = {};

  int nkb = K >> 5;

  // ---- tile copy: A 64x32 (2 rounds), W 128x32 (4 rounds), 16B per lane ----
  auto issue_tiles = [&](int kb, int buf) {
#pragma unroll
    for (int r = 0; r < 2; ++r) {
      int idx = r * 256 + tid;
      int row = idx >> 3, seg = idx & 7;
      unsigned gofs = (unsigned)(((rowBase + row) * K + kb * 32 + seg * 4) * 4);
      unsigned lofs = ldsA[buf] + (unsigned)(row * (LPAD * 4) + seg * 16);
      async_copy_b128(lofs, gofs, aBase);
    }
#pragma unroll
    for (int r = 0; r < 4; ++r) {
      int idx = r * 256 + tid;
      int row = idx >> 3, seg = idx & 7;
      unsigned gofs = (unsigned)(((colBase + row) * K + kb * 32 + seg * 4) * 4);
      unsigned lofs = ldsW[buf] + (unsigned)(row * (LPAD * 4) + seg * 16);
      async_copy_b128(lofs, gofs, wBase);
    }
  };

  issue_tiles(0, 0);
  wait_async0();
  __syncthreads();

  for (int kb = 0; kb < nkb; ++kb) {
    int cur = kb & 1;
    if (kb + 1 < nkb) issue_tiles(kb + 1, cur ^ 1);   // overlap copy & math
#pragma unroll
    for (int kk = 0; kk < 8; ++kk) {
      v2f a = *reinterpret_cast<const v2f*>(&sA[cur][wm + lm][kk * 4 + 2 * half]);
#pragma unroll
      for (int t = 0; t < 4; ++t) {
        v2f bv =
            *reinterpret_cast<const v2f*>(&sW[cur][wn + t * 16 + lm][kk * 4 + 2 * half]);
        acc[t] = wmma4(a, bv, acc[t]);
      }
    }
    wait_async0();
    __syncthreads();
  }

#pragma unroll
  for (int t = 0; t < 4; ++t) {
    int n = colBase + wn + t * 16 + lm;
    float bb = bias[n];
#pragma unroll
    for (int r = 0; r < 8; ++r) {
      int m = rowBase + wm + r + 8 * half;
      float v = acc[t][r] + bb;
      if (relu) v = fmaxf(v, 0.f);
      C[(size_t)m * N + n] = v;
    }
  }
}

// ---------------- attention: one block per (b,h) -----------------------------
__global__ __launch_bounds__(256)
void attention_kernel(const float* __restrict__ qkv, float* __restrict__ o) {
  __shared__ __align__(16) float s[64][66];   // 66 keeps float2 reads 8B-aligned
  int bh = blockIdx.x;
  int b = bh >> 3;
  int h = bh & 7;
  int tid = threadIdx.x, lane = tid & 31, wave = tid >> 5;
  int lm = lane & 15, half = lane >> 4;
  const float scale = 0.17677669529663688f;   // 1/sqrt(32)
  const float* base = qkv + (size_t)b * 64 * (3 * Dm);

  // phase 1: S = scale * Q K^T  (4x2 tiles per-wave -> 64x64)
  {
    int sm = (wave & 3) * 16;
    int sn = (wave >> 2) * 32;
    v8f a0 = {}, a1 = {};
    const float* qrow = base + (size_t)(sm + lm) * (3 * Dm) + h * 32 + 2 * half;
    const float* k0r =
        base + (size_t)(sn + lm) * (3 * Dm) + Dm + h * 32 + 2 * half;
    const float* k1r =
        base + (size_t)(sn + 16 + lm) * (3 * Dm) + Dm + h * 32 + 2 * half;
#pragma unroll
    for (int kk = 0; kk < 8; ++kk) {
      v2f a = *reinterpret_cast<const v2f*>(qrow + kk * 4);
      a0 = wmma4(a, *reinterpret_cast<const v2f*>(k0r + kk * 4), a0);
      a1 = wmma4(a, *reinterpret_cast<const v2f*>(k1r + kk * 4), a1);
    }
#pragma unroll
    for (int r = 0; r < 8; ++r) {
      int row = sm + r + 8 * half;
      s[row][sn + lm]      = a0[r] * scale;
      s[row][sn + 16 + lm] = a1[r] * scale;
    }
  }
  __syncthreads();

  // phase 2: row softmax (threads 0..63, one row each)
  if (tid < 64) {
    float m = -1e30f;
    for (int c = 0; c < 64; ++c) m = fmaxf(m, s[tid][c]);
    float sum = 0.f;
    for (int c = 0; c < 64; ++c) {
      float e = __expf(s[tid][c] - m);
      s[tid][c] = e;
      sum += e;
    }
    float inv = 1.f / sum;
    for (int c = 0; c < 64; ++c) s[tid][c] *= inv;
  }
  __syncthreads();

  // phase 3: O = P V  (8 tiles of 16x16 -> 64x32), P frags from LDS
  {
    int om = (wave & 3) * 16;
    int on = (wave >> 2) * 16;
    v8f acc = {};
#pragma unroll
    for (int kk = 0; kk < 16; ++kk) {
      int k = kk * 4 + 2 * half;
      v2f a = *reinterpret_cast<const v2f*>(&s[om + lm][k]);
      v2f bv;
      bv.x = base[(size_t)k * (3 * Dm) + 2 * Dm + h * 32 + on + lm];
      bv.y = base[(size_t)(k + 1) * (3 * Dm) + 2 * Dm + h * 32 + on + lm];
      acc = wmma4(a, bv, acc);
    }
#pragma unroll
    for (int r = 0; r < 8; ++r) {
      int row = om + r + 8 * half;
      o[(size_t)(b * 64 + row) * Dm + h * 32 + on + lm] = acc[r];
    }
  }
}

// ---------------- residual + LayerNorm: out = LN(x + d) ----------------------
__global__ __launch_bounds__(256)
void add_ln_kernel(const float* __restrict__ x, const float* __restrict__ d,
                   const float* __restrict__ g, const float* __restrict__ bb,
                   float* __restrict__ out) {
  __shared__ float rs[256];
  __shared__ float rq[256];
  int tok = blockIdx.x, c = threadIdx.x;
  float v = x[(size_t)tok * Dm + c] + d[(size_t)tok * Dm + c];
  rs[c] = v;
  rq[c] = v * v;
  __syncthreads();
  for (int st = 128; st > 0; st >>= 1) {
    if (c < st) {
      rs[c] += rs[c + st];
      rq[c] += rq[c + st];
    }
    __syncthreads();
  }
  float mean = rs[0] * (1.f / Dm);
  float var  = rq[0] * (1.f / Dm) - mean * mean;
  out[(size_t)tok * Dm + c] = (v - mean) * rsqrtf(var + EPSV) * g[c] + bb[c];
}

// ----------------------------------------------------------------------------
extern "C" void kernel_launch(void* const* d_in, const int* in_sizes, int n_in,
                              void* d_out, int out_size, void* d_ws,
                              size_t ws_size, hipStream_t stream) {
  (void)in_sizes; (void)n_in; (void)out_size; (void)ws_size;
  const int*   word_idx  = (const int*)d_in[0];
  const int*   ngram_idx = (const int*)d_in[1];
  const float* fc1_w = (const float*)d_in[2];
  const float* fc1_b = (const float*)d_in[3];
  const float* fc2_w = (const float*)d_in[4];
  const float* fc2_b = (const float*)d_in[5];
  const float* qkv_w = (const float*)d_in[6];
  const float* qkv_b = (const float*)d_in[7];
  const float* out_w = (const float*)d_in[8];
  const float* out_b = (const float*)d_in[9];
  const float* ln1_g = (const float*)d_in[10];
  const float* ln1_b = (const float*)d_in[11];
  const float* ff1_w = (const float*)d_in[12];
  const float* ff1_b = (const float*)d_in[13];
  const float* ff2_w = (const float*)d_in[14];
  const float* ff2_b = (const float*)d_in[15];
  const float* ln2_g = (const float*)d_in[16];
  const float* ln2_b = (const float*)d_in[17];

  float* ws   = (float*)d_ws;
  float* h0   = ws;                              // 2048*512
  float* x    = h0   + (size_t)NTOK * FFDN;      // 2048*256
  float* qkvb = x    + (size_t)NTOK * Dm;        // 2048*768
  float* attn = qkvb + (size_t)NTOK * 3 * Dm;    // 2048*256
  float* tmp  = attn + (size_t)NTOK * Dm;        // 2048*256
  float* ffh  = tmp  + (size_t)NTOK * Dm;        // 2048*1024

  gather_fc1_kernel<<<NTOK, 256, 0, stream>>>(word_idx, ngram_idx, fc1_w,
                                              fc1_b, h0);
  // x = h0 @ fc2_w^T + fc2_b   (M=2048, N=256, K=512)
  gemm_wmma_kernel<<<dim3(NTOK / 64, 256 / 128), 256, 0, stream>>>(
      h0, fc2_w, fc2_b, x, NTOK, 256, 512, 0);

  for (int i = 0; i < NLAYERS; ++i) {
    gemm_wmma_kernel<<<dim3(NTOK / 64, 768 / 128), 256, 0, stream>>>(
        x, qkv_w + (size_t)i * 768 * 256, qkv_b + i * 768, qkvb, NTOK, 768,
        256, 0);
    attention_kernel<<<Bsz * Hh, 256, 0, stream>>>(qkvb, attn);
    gemm_wmma_kernel<<<dim3(NTOK / 64, 256 / 128), 256, 0, stream>>>(
        attn, out_w + (size_t)i * 256 * 256, out_b + i * 256, tmp, NTOK, 256,
        256, 0);
    add_ln_kernel<<<NTOK, 256, 0, stream>>>(x, tmp, ln1_g + i * 256,
                                            ln1_b + i * 256, x);
    gemm_wmma_kernel<<<dim3(NTOK / 64, 1024 / 128), 256, 0, stream>>>(
        x, ff1_w + (size_t)i * 1024 * 256, ff1_b + i * 1024, ffh, NTOK, 1024,
        256, 1);
    gemm_wmma_kernel<<<dim3(NTOK / 64, 256 / 128), 256, 0, stream>>>(
        ffh, ff2_w + (size_t)i * 256 * 1024, ff2_b + i * 256, tmp, NTOK, 256,
        1024, 0);
    float* dst = (i == NLAYERS - 1) ? (float*)d_out : x;
    add_ln_kernel<<<NTOK, 256, 0, stream>>>(x, tmp, ln2_g + i * 256,
                                            ln2_b + i * 256, dst);
  }
}